// GAT_5428838662453
// MI455X (gfx1250) — compile-verified
//
#include <hip/hip_runtime.h>
#include <hip/hip_bf16.h>

// ---------------- problem constants ----------------
#define NN      50000
#define EE      800000
#define ET      (EE + NN)          // edges incl. self loops
#define F_IN    256
#define HID     64
#define HEADS   4
#define HC      (HEADS * HID)      // 256
#define CLASSES 32
#define NEG_SLOPE 0.2f
#define KDIM    256                // K for both GEMMs

typedef float v2f __attribute__((ext_vector_type(2)));
typedef float v8f __attribute__((ext_vector_type(8)));

// ---- ordered-uint encoding for float atomicMax (monotonic) ----
__device__ __forceinline__ unsigned fenc(float f) {
    unsigned u = __float_as_uint(f);
    return (u & 0x80000000u) ? ~u : (u | 0x80000000u);
}
__device__ __forceinline__ float fdec(unsigned u) {
    return (u & 0x80000000u) ? __uint_as_float(u & 0x7FFFFFFFu)
                             : __uint_as_float(~u);
}
__device__ __forceinline__ float leaky(float x) {
    return x > 0.f ? x : NEG_SLOPE * x;
}

// =====================================================================
// GEMM: C[M x NOUT] = A[M x 256] * B[256 x NOUT], f32 WMMA 16x16x4.
// Block = 4 waves sharing one 16*NSUB-column B strip staged in LDS
// (64-row chunks, row stride padded +8 dwords -> half-waves 16 banks
// apart, conflict-free). Each wave computes a 16 x (16*NSUB) tile.
// =====================================================================
template <int NSUB, int NOUT>
__global__ __launch_bounds__(128)
void gat_gemm_wmma(const float* __restrict__ A, const float* __restrict__ B,
                   float* __restrict__ C, int M) {
    constexpr int STRIPS = NOUT / (16 * NSUB);
    constexpr int COLS   = 16 * NSUB;       // 64 or 32
    constexpr int STRIDE = COLS + 8;        // LDS row stride (dwords)
    constexpr int CHUNK  = 64;              // K rows staged per round
    constexpr int NF4    = CHUNK * COLS / 4;        // float4s per chunk
    constexpr int F4PT   = NF4 / 128;               // float4s per thread

    __shared__ float bl[CHUNK * STRIDE];

    const int lane  = threadIdx.x & 31;
    const int wave  = threadIdx.x >> 5;
    const int strip = blockIdx.x % STRIPS;
    const int mtile = (blockIdx.x / STRIPS) * 4 + wave;
    const int n0    = strip * COLS;
    const bool valid = (mtile * 16 < M);

    const int l15   = lane & 15;
    const int khalf = lane >> 4;            // 0: K=k0,k0+1  1: K=k0+2,k0+3
    const float* __restrict__ arow =
        A + (long)(valid ? mtile * 16 + l15 : l15) * KDIM;

    v8f acc[NSUB];
#pragma unroll
    for (int s = 0; s < NSUB; ++s) acc[s] = (v8f){0.f,0.f,0.f,0.f,0.f,0.f,0.f,0.f};

    for (int kc = 0; kc < KDIM; kc += CHUNK) {
        __syncthreads();                    // LDS reuse guard
        // cooperative, coalesced float4 staging of B[kc:kc+64, n0:n0+COLS]
#pragma unroll
        for (int i = 0; i < F4PT; ++i) {
            const int j  = threadIdx.x + i * 128;   // float4 index in chunk
            const int r  = j / (COLS / 4);
            const int c4 = j % (COLS / 4);
            const float4 v =
                *(const float4*)(B + (long)(kc + r) * NOUT + n0 + c4 * 4);
            *(float4*)&bl[r * STRIDE + c4 * 4] = v;
        }
        __syncthreads();

#pragma unroll
        for (int kk = 0; kk < CHUNK; kk += 4) {
            const int kb = kc + kk + khalf * 2;
            v2f a = *(const v2f*)(arow + kb);       // 8B-aligned (kb even)
            const int rl = kk + khalf * 2;          // local LDS row
#pragma unroll
            for (int s = 0; s < NSUB; ++s) {
                const int cl = s * 16 + l15;
                v2f b;
                b.x = bl[(rl + 0) * STRIDE + cl];
                b.y = bl[(rl + 1) * STRIDE + cl];
                acc[s] = __builtin_amdgcn_wmma_f32_16x16x4_f32(
                    false, a, false, b, (short)0, acc[s], false, false);
            }
        }
    }

    if (!valid) return;
    // C/D layout: VGPR v -> row m0 + 8*khalf + v, col n0 + (lane&15)
#pragma unroll
    for (int s = 0; s < NSUB; ++s) {
        const int col = n0 + s * 16 + l15;
#pragma unroll
        for (int v = 0; v < 8; ++v) {
            const int r = mtile * 16 + khalf * 8 + v;
            C[(long)r * NOUT + col] = acc[s][v];
        }
    }
}

// =====================================================================
// Per-node attention logits: al_s[n,h] = sum_c h[n,h,c]*a_src[h,c]
// =====================================================================
template <int H, int C>
__global__ void gat_logits(const float* __restrict__ h,
                           const float* __restrict__ a_src,
                           const float* __restrict__ a_dst,
                           float* __restrict__ al_s, float* __restrict__ al_d) {
    const int t = blockIdx.x * blockDim.x + threadIdx.x;
    if (t >= NN * H) return;
    const int n = t / H, hh = t % H;
    const float4* hr = (const float4*)(h + (long)n * H * C + hh * C);
    const float4* as = (const float4*)(a_src + hh * C);
    const float4* ad = (const float4*)(a_dst + hh * C);
    float ss = 0.f, sd = 0.f;
#pragma unroll
    for (int c = 0; c < C / 4; ++c) {
        const float4 v = hr[c], s4 = as[c], d4 = ad[c];
        ss += v.x * s4.x + v.y * s4.y + v.z * s4.z + v.w * s4.w;
        sd += v.x * d4.x + v.y * d4.y + v.z * d4.z + v.w * d4.w;
    }
    al_s[t] = ss;
    al_d[t] = sd;
}

// =====================================================================
// Pass 1 over edges: segment max of leaky(al_s[src]+al_dst[dst]) per head
// =====================================================================
template <int H>
__global__ void gat_edge_max(const int* __restrict__ ei,
                             const float* __restrict__ al_s,
                             const float* __restrict__ al_d,
                             unsigned* __restrict__ menc) {
    const int t = blockIdx.x * blockDim.x + threadIdx.x;
    if (t >= ET) return;
    int s, d;
    if (t < EE) { s = ei[t]; d = ei[EE + t]; } else { s = d = t - EE; }
#pragma unroll
    for (int hh = 0; hh < H; ++hh) {
        float e = leaky(al_s[s * H + hh] + al_d[d * H + hh]);
        atomicMax(&menc[d * H + hh], fenc(e));
    }
}

// =====================================================================
// Pass 2, layer 1 (H=4, C=64): one wave per edge. float4 gather of
// h[src], then denom[dst,h] += ex_h ; acc[dst,:] += ex_h * h[src,:]
// =====================================================================
__global__ __launch_bounds__(256)
void gat_edge_acc1(const int* __restrict__ ei,
                   const float* __restrict__ al_s, const float* __restrict__ al_d,
                   const unsigned* __restrict__ menc,
                   const float* __restrict__ h,
                   float* __restrict__ denom, float* __restrict__ acc) {
    const int lane = threadIdx.x & 31;
    const int wid  = (blockIdx.x * blockDim.x + threadIdx.x) >> 5;
    if (wid >= ET) return;
    int s, d;
    if (wid < EE) { s = ei[wid]; d = ei[EE + wid]; } else { s = d = wid - EE; }

    float ex[HEADS];
#pragma unroll
    for (int hh = 0; hh < HEADS; ++hh) {
        float e = leaky(al_s[s * HEADS + hh] + al_d[d * HEADS + hh]);
        ex[hh] = __expf(e - fdec(menc[d * HEADS + hh]));
    }
    if (lane < HEADS) atomicAdd(&denom[d * HEADS + lane], ex[lane]);

    const float4* __restrict__ hs4 = (const float4*)(h + (long)s * HC);
    float* __restrict__        ad  = acc + (long)d * HC;
#pragma unroll
    for (int i = 0; i < 2; ++i) {            // 2 x 128B-coalesced gathers
        const int q  = i * 32 + lane;        // float4 index 0..63
        const float4 v = hs4[q];
        const int c0 = q * 4;                // 4 channels share one head
        const float w = ex[c0 >> 6];
        atomicAdd(&ad[c0 + 0], w * v.x);
        atomicAdd(&ad[c0 + 1], w * v.y);
        atomicAdd(&ad[c0 + 2], w * v.z);
        atomicAdd(&ad[c0 + 3], w * v.w);
    }
}

// =====================================================================
// Pass 2, layer 2 (H=1, C=32): one wave per edge, lane == channel
// =====================================================================
__global__ __launch_bounds__(256)
void gat_edge_acc2(const int* __restrict__ ei,
                   const float* __restrict__ al_s, const float* __restrict__ al_d,
                   const unsigned* __restrict__ menc,
                   const float* __restrict__ h,
                   float* __restrict__ denom, float* __restrict__ acc) {
    const int lane = threadIdx.x & 31;
    const int wid  = (blockIdx.x * blockDim.x + threadIdx.x) >> 5;
    if (wid >= ET) return;
    int s, d;
    if (wid < EE) { s = ei[wid]; d = ei[EE + wid]; } else { s = d = wid - EE; }

    float e  = leaky(al_s[s] + al_d[d]);
    float ex = __expf(e - fdec(menc[d]));
    if (lane == 0) atomicAdd(&denom[d], ex);
    atomicAdd(&acc[(long)d * CLASSES + lane], ex * h[(long)s * CLASSES + lane]);
}

// =====================================================================
// Layer-1 finalize: h1 = ELU(acc/denom + b1)   (overwrites hmid buffer)
// =====================================================================
__global__ void gat_finalize1(const float* __restrict__ acc,
                              const float* __restrict__ denom,
                              const float* __restrict__ b1,
                              float* __restrict__ h1) {
    const int t = blockIdx.x * blockDim.x + threadIdx.x;   // float4 index
    if (t >= NN * (HC / 4)) return;
    const int n = t >> 6, q = t & 63;
    const int c0 = q * 4;                    // 4 channels share one head
    const float dn = denom[n * HEADS + (c0 >> 6)];
    const float4 a = ((const float4*)acc)[t];
    const float4 b = ((const float4*)b1)[q];
    float4 o;
    o.x = a.x / dn + b.x; o.y = a.y / dn + b.y;
    o.z = a.z / dn + b.z; o.w = a.w / dn + b.w;
    o.x = o.x > 0.f ? o.x : (__expf(o.x) - 1.f);
    o.y = o.y > 0.f ? o.y : (__expf(o.y) - 1.f);
    o.z = o.z > 0.f ? o.z : (__expf(o.z) - 1.f);
    o.w = o.w > 0.f ? o.w : (__expf(o.w) - 1.f);
    ((float4*)h1)[t] = o;
}

// =====================================================================
// Layer-2 finalize + log_softmax over 32 classes: one wave per node
// =====================================================================
__global__ __launch_bounds__(256)
void gat_finalize2(const float* __restrict__ acc,
                   const float* __restrict__ denom,
                   const float* __restrict__ b2,
                   float* __restrict__ out) {
    const int lane = threadIdx.x & 31;             // lane == class
    const int n    = (blockIdx.x * blockDim.x + threadIdx.x) >> 5;
    if (n >= NN) return;
    float v = acc[(long)n * CLASSES + lane] / denom[n] + b2[lane];

    float mx = v;
#pragma unroll
    for (int o = 16; o > 0; o >>= 1) mx = fmaxf(mx, __shfl_xor(mx, o, 32));
    float ex = __expf(v - mx), sum = ex;
#pragma unroll
    for (int o = 16; o > 0; o >>= 1) sum += __shfl_xor(sum, o, 32);
    out[(long)n * CLASSES + lane] = v - mx - __logf(sum);
}

// =====================================================================
// host-side launcher
// =====================================================================
static inline size_t alignup(size_t x) { return (x + 255) & ~(size_t)255; }

extern "C" void kernel_launch(void* const* d_in, const int* in_sizes, int n_in,
                              void* d_out, int out_size, void* d_ws, size_t ws_size,
                              hipStream_t stream) {
    const float* x      = (const float*)d_in[0];
    const int*   ei     = (const int*)  d_in[1];   // [2,E] flat: src then dst
    const float* W1     = (const float*)d_in[2];
    const float* a_src1 = (const float*)d_in[3];
    const float* a_dst1 = (const float*)d_in[4];
    const float* b1     = (const float*)d_in[5];
    const float* W2     = (const float*)d_in[6];
    const float* a_src2 = (const float*)d_in[7];
    const float* a_dst2 = (const float*)d_in[8];
    const float* b2     = (const float*)d_in[9];
    float* out = (float*)d_out;

    // ---- workspace carve-up ----
    char* p = (char*)d_ws;
    auto take = [&](size_t bytes) { char* q = p; p += alignup(bytes); return q; };
    float*    hmid   = (float*)   take((size_t)NN * HC * 4);      // x@W1, later h1
    float*    acc1   = (float*)   take((size_t)NN * HC * 4);
    float*    al_s1  = (float*)   take((size_t)NN * HEADS * 4);
    float*    al_d1  = (float*)   take((size_t)NN * HEADS * 4);
    unsigned* m1     = (unsigned*)take((size_t)NN * HEADS * 4);
    float*    den1   = (float*)   take((size_t)NN * HEADS * 4);
    float*    h2     = (float*)   take((size_t)NN * CLASSES * 4); // h1@W2
    float*    acc2   = (float*)   take((size_t)NN * CLASSES * 4);
    float*    al_s2  = (float*)   take((size_t)NN * 4);
    float*    al_d2  = (float*)   take((size_t)NN * 4);
    unsigned* m2     = (unsigned*)take((size_t)NN * 4);
    float*    den2   = (float*)   take((size_t)NN * 4);

    // ---- zero accumulators (graph-capturable async memsets) ----
    hipMemsetAsync(acc1, 0, (size_t)NN * HC * 4, stream);
    hipMemsetAsync(m1,   0, (size_t)NN * HEADS * 4, stream);   // fdec(0) sentinel
    hipMemsetAsync(den1, 0, (size_t)NN * HEADS * 4, stream);
    hipMemsetAsync(acc2, 0, (size_t)NN * CLASSES * 4, stream);
    hipMemsetAsync(m2,   0, (size_t)NN * 4, stream);
    hipMemsetAsync(den2, 0, (size_t)NN * 4, stream);

    const int MT = NN / 16;                   // 3125 M-tiles
    // ---------------- layer 1 ----------------
    // hmid = x @ W1 : 4 strips x ceil(3125/4) blocks of 4 waves
    gat_gemm_wmma<4, HC><<<4 * ((MT + 3) / 4), 128, 0, stream>>>(x, W1, hmid, NN);
    gat_logits<HEADS, HID><<<(NN * HEADS + 255) / 256, 256, 0, stream>>>(
        hmid, a_src1, a_dst1, al_s1, al_d1);
    gat_edge_max<HEADS><<<(ET + 255) / 256, 256, 0, stream>>>(ei, al_s1, al_d1, m1);
    gat_edge_acc1<<<(ET + 7) / 8, 256, 0, stream>>>(ei, al_s1, al_d1, m1, hmid,
                                                    den1, acc1);
    gat_finalize1<<<(NN * (HC / 4) + 255) / 256, 256, 0, stream>>>(
        acc1, den1, b1, hmid);

    // ---------------- layer 2 ----------------
    // h2 = h1 @ W2 : 1 strip x ceil(3125/4) blocks
    gat_gemm_wmma<2, CLASSES><<<(MT + 3) / 4, 128, 0, stream>>>(hmid, W2, h2, NN);
    gat_logits<1, CLASSES><<<(NN + 255) / 256, 256, 0, stream>>>(
        h2, a_src2, a_dst2, al_s2, al_d2);
    gat_edge_max<1><<<(ET + 255) / 256, 256, 0, stream>>>(ei, al_s2, al_d2, m2);
    gat_edge_acc2<<<(ET + 7) / 8, 256, 0, stream>>>(ei, al_s2, al_d2, m2, h2,
                                                    den2, acc2);
    gat_finalize2<<<(NN + 7) / 8, 256, 0, stream>>>(acc2, den2, b2, out);
}